// GINEConv_88364657148500
// MI455X (gfx1250) — compile-verified
//
#include <hip/hip_runtime.h>
#include <hip/hip_bf16.h>

typedef __attribute__((ext_vector_type(2))) float v2f;
typedef __attribute__((ext_vector_type(8))) float v8f;

#define DIM 64

// ---------------------------------------------------------------------------
// Kernel 1: zero the aggregation workspace
// ---------------------------------------------------------------------------
__global__ void gine_zero(float* __restrict__ p, long long n) {
    long long i = (long long)blockIdx.x * blockDim.x + threadIdx.x;
    if (i < n) p[i] = 0.0f;
}

// ---------------------------------------------------------------------------
// Kernel 2: edge gather + relu + scatter-add (atomic f32 into L2-resident agg)
// 32 lanes per edge, each lane handles 2 channels (float2 gather, 2 atomics).
// ---------------------------------------------------------------------------
__global__ void gine_edge(const float* __restrict__ x,
                          const long long* __restrict__ ei,
                          float* __restrict__ agg,
                          long long E) {
    long long t = (long long)blockIdx.x * blockDim.x + threadIdx.x;
    long long e = t >> 5;
    int lane = (int)(t & 31);
    if (e >= E) return;
    long long s = ei[e];
    long long d = ei[E + e];
    const float2* xs = (const float2*)(x + s * DIM);
    float2 v = xs[lane];
    float a0 = v.x > 0.0f ? v.x : 0.0f;
    float a1 = v.y > 0.0f ? v.y : 0.0f;
    float* dp = agg + d * DIM + (long long)lane * 2;
    atomicAdd(dp + 0, a0);
    atomicAdd(dp + 1, a1);
}

// ---------------------------------------------------------------------------
// Kernel 3: h = x + agg ; out = relu( relu(h@W1+b1) @ W2 + b2 )
// 8 waves per block, each wave computes a 16x64 output row-tile via
// v_wmma_f32_16x16x4_f32 (exact fp32).
// ---------------------------------------------------------------------------
__global__ void __launch_bounds__(256)
gine_mlp_wmma(const float* __restrict__ x,
              const float* __restrict__ agg,
              const float* __restrict__ W1, const float* __restrict__ b1,
              const float* __restrict__ W2, const float* __restrict__ b2,
              float* __restrict__ out, int N) {
    __shared__ float sW1[DIM * DIM];          // 16 KB
    __shared__ float sW2[DIM * DIM];          // 16 KB
    __shared__ float sh[8][16 * DIM];         // 32 KB: per-wave 16x64 h tile

    const int tid = threadIdx.x;

    // Cooperative stage of both weight matrices into LDS.
    #pragma unroll
    for (int i = tid; i < DIM * DIM; i += 256) {
        sW1[i] = W1[i];
        sW2[i] = W2[i];
    }
    __syncthreads();

    const int wave   = tid >> 5;
    const int lane   = tid & 31;
    const int lane16 = lane & 15;
    const int half   = lane >> 4;            // 0: lanes 0-15, 1: lanes 16-31
    const long long R = ((long long)blockIdx.x * 8 + wave) * 16; // tile row base

    float* hT = &sh[wave][0];

    // --- Load h tile = x + agg (16 rows x 64 cols), clamped rows for tail ---
    #pragma unroll
    for (int j = 0; j < 8; ++j) {
        int v4  = j * 32 + lane;             // float4 index within tile (256 total)
        int m   = v4 >> 4;                   // row in tile
        int c4  = v4 & 15;                   // float4-column
        long long row = R + m;
        if (row >= N) row = N - 1;           // clamped rows feed dead outputs only
        long long g = row * DIM + (long long)c4 * 4;
        float4 xv = *(const float4*)(x + g);
        float4 av = *(const float4*)(agg + g);
        float4 hv;
        hv.x = xv.x + av.x; hv.y = xv.y + av.y;
        hv.z = xv.z + av.z; hv.w = xv.w + av.w;
        *(float4*)(hT + (long long)v4 * 4) = hv;
    }

    v8f acc[4];

    // ================= Layer 1: h @ W1 + b1, relu =================
    #pragma unroll
    for (int nt = 0; nt < 4; ++nt) {
        float bv = b1[nt * 16 + lane16];
        #pragma unroll
        for (int r = 0; r < 8; ++r) acc[nt][r] = bv;
    }
    #pragma unroll
    for (int kb = 0; kb < DIM; kb += 4) {
        int k0 = kb + 2 * half;
        v2f a = *(const v2f*)(hT + lane16 * DIM + k0);   // A frag: row=lane16
        #pragma unroll
        for (int nt = 0; nt < 4; ++nt) {
            v2f b;
            b.x = sW1[(k0 + 0) * DIM + nt * 16 + lane16];
            b.y = sW1[(k0 + 1) * DIM + nt * 16 + lane16];
            acc[nt] = __builtin_amdgcn_wmma_f32_16x16x4_f32(
                false, a, false, b, (short)0, acc[nt], false, false);
        }
    }
    // relu -> write h1 back into the same per-wave LDS tile (same-wave only)
    #pragma unroll
    for (int nt = 0; nt < 4; ++nt) {
        #pragma unroll
        for (int r = 0; r < 8; ++r) {
            float v = acc[nt][r];
            v = v > 0.0f ? v : 0.0f;
            hT[(r + 8 * half) * DIM + nt * 16 + lane16] = v;
        }
    }

    // ================= Layer 2: h1 @ W2 + b2, relu, store =================
    #pragma unroll
    for (int nt = 0; nt < 4; ++nt) {
        float bv = b2[nt * 16 + lane16];
        #pragma unroll
        for (int r = 0; r < 8; ++r) acc[nt][r] = bv;
    }
    #pragma unroll
    for (int kb = 0; kb < DIM; kb += 4) {
        int k0 = kb + 2 * half;
        v2f a = *(const v2f*)(hT + lane16 * DIM + k0);
        #pragma unroll
        for (int nt = 0; nt < 4; ++nt) {
            v2f b;
            b.x = sW2[(k0 + 0) * DIM + nt * 16 + lane16];
            b.y = sW2[(k0 + 1) * DIM + nt * 16 + lane16];
            acc[nt] = __builtin_amdgcn_wmma_f32_16x16x4_f32(
                false, a, false, b, (short)0, acc[nt], false, false);
        }
    }
    #pragma unroll
    for (int nt = 0; nt < 4; ++nt) {
        #pragma unroll
        for (int r = 0; r < 8; ++r) {
            long long row = R + r + 8 * half;
            if (row < N) {
                float v = acc[nt][r];
                out[row * DIM + nt * 16 + lane16] = v > 0.0f ? v : 0.0f;
            }
        }
    }
}

// ---------------------------------------------------------------------------
extern "C" void kernel_launch(void* const* d_in, const int* in_sizes, int n_in,
                              void* d_out, int out_size, void* d_ws, size_t ws_size,
                              hipStream_t stream) {
    const float*     x  = (const float*)d_in[0];
    const long long* ei = (const long long*)d_in[1];   // int64 edge_index [2, E]
    const float*     W1 = (const float*)d_in[2];
    const float*     b1 = (const float*)d_in[3];
    const float*     W2 = (const float*)d_in[4];
    const float*     b2 = (const float*)d_in[5];
    float*           out = (float*)d_out;

    const long long N = (long long)in_sizes[0] / DIM;
    const long long E = (long long)in_sizes[1] / 2;

    float* agg = (float*)d_ws;                 // N*DIM floats of scratch
    const long long nAgg = N * DIM;

    // 1) zero agg
    {
        long long nb = (nAgg + 255) / 256;
        gine_zero<<<(unsigned)nb, 256, 0, stream>>>(agg, nAgg);
    }
    // 2) edge gather + relu + atomic scatter-add
    {
        long long threads = E * 32;
        long long nb = (threads + 255) / 256;
        gine_edge<<<(unsigned)nb, 256, 0, stream>>>(x, ei, agg, E);
    }
    // 3) fused h = x + agg, MLP via WMMA, outer relu
    {
        int nTiles = (int)((N + 127) / 128);   // 128 rows per block (8 waves x 16)
        gine_mlp_wmma<<<nTiles, 256, 0, stream>>>(x, agg, W1, b1, W2, b2, out, (int)N);
    }
}